// AI4SWE_49323404427329
// MI455X (gfx1250) — compile-verified
//
#include <hip/hip_runtime.h>
#include <stdint.h>

#define N_CELLS 8388608
#define N_HALO  1048576
#define N_QOUT  11
#define BLOCK   256

// ---- LDS byte-offset helper: generic -> addrspace(3) ptr -> 32-bit offset ----
typedef __attribute__((address_space(3))) char lds_char_t;

static __device__ __forceinline__ uint32_t lds_byte_off(const void* p) {
    return (uint32_t)(uintptr_t)(lds_char_t*)p;
}

// ---- CDNA5 async global->LDS copy (ASYNCcnt-tracked), GVS addressing ----
static __device__ __forceinline__ void async_copy_b64(uint32_t lds_off,
                                                      uint32_t byte_off,
                                                      const void* base) {
    asm volatile("global_load_async_to_lds_b64 %0, %1, %2"
                 :
                 : "v"(lds_off), "v"(byte_off), "s"(base)
                 : "memory");
}

static __device__ __forceinline__ void wait_asynccnt0() {
    asm volatile("s_wait_asynccnt 0" ::: "memory");
}

// Pin a uniform pointer in an SGPR pair and make it opaque to the optimizer,
// so each access stays gep(uniform_base, zext(divergent_u32)) and the backend
// selects the global_load/store saddr + 32-bit voffset form (no per-lane
// 64-bit address chains for the 32MB/4MB field strides).
static __device__ __forceinline__ const float* sgpr_base(const float* p) {
    asm("" : "+s"(p));
    return p;
}
static __device__ __forceinline__ float* sgpr_base_mut(float* p) {
    asm("" : "+s"(p));
    return p;
}

// Gather 12 raw fields at cell index `c` and form the 11 derived quantities.
static __device__ __forceinline__ void derive(const float* const __restrict__ b[12],
                                              uint32_t c, float d[N_QOUT]) {
    float f0  = b[0][c];    // u
    float f1  = b[1][c];    // v
    float f2  = b[2][c];    // b_u
    float f3  = b[3][c];    // b_v
    float f4  = b[4][c];    // h
    float f5  = b[5][c];    // Hb
    float f6  = b[6][c];    // hh
    float f7  = b[7][c];    // dif_h
    float f8  = b[8][c];    // eta1
    float f9  = b[9][c];    // k_u
    float f10 = b[10][c];   // k_v
    float f11 = b[11][c];   // k3
    d[0]  = f0;
    d[1]  = f1;
    d[2]  = f2;
    d[3]  = f3;
    d[4]  = f4;
    d[5]  = f6;
    d[6]  = f7;
    d[7]  = f4 + f5;
    d[8]  = f8;
    d[9]  = fminf(f9,  f11);
    d[10] = fminf(f10, f11);
}

__global__ __launch_bounds__(BLOCK)
void AI4SWE_49323404427329_halo_gather(const float* __restrict__ fields,
                                       const int*   __restrict__ src_idx,
                                       const float* __restrict__ weights,
                                       float*       __restrict__ out) {
    __shared__ int2   sIdx[BLOCK];
    __shared__ float2 sW[BLOCK];

    const int tid = threadIdx.x;
    const int e   = blockIdx.x * BLOCK + tid;

    // Stream per-halo metadata straight into LDS via the async path (8B/lane each).
    const uint32_t boff = (uint32_t)e * 8u;   // both arrays are 8 bytes per halo
    async_copy_b64(lds_byte_off(&sIdx[tid]), boff, src_idx);
    async_copy_b64(lds_byte_off(&sW[tid]),   boff, weights);

    // Uniform per-field bases pinned in SGPR pairs (opaque: no reassociation).
    const float* b[12];
    #pragma unroll
    for (int q = 0; q < 12; ++q) b[q] = sgpr_base(fields + (size_t)q * N_CELLS);

    // Uniform per-row output bases, same treatment.
    float* ob[N_QOUT];
    #pragma unroll
    for (int q = 0; q < N_QOUT; ++q) ob[q] = sgpr_base_mut(out + (size_t)q * N_HALO);

    wait_asynccnt0();                          // ASYNCcnt==0 -> LDS writes landed

    const int2   idx = sIdx[tid];              // ds_load of own slot only
    const float2 w   = sW[tid];

    float acc[N_QOUT];
    {
        float d[N_QOUT];
        derive(b, (uint32_t)idx.x, d);
        #pragma unroll
        for (int q = 0; q < N_QOUT; ++q) acc[q] = w.x * d[q];
    }
    // ~50% of halos have w.y == 0.0 exactly (weights are {1,0} or {.5,.5});
    // masked-off lanes issue no gather requests -> ~25% less random traffic.
    if (w.y != 0.0f) {
        float d[N_QOUT];
        derive(b, (uint32_t)idx.y, d);
        #pragma unroll
        for (int q = 0; q < N_QOUT; ++q) acc[q] = fmaf(w.y, d[q], acc[q]);
    }

    // Streaming output: non-temporal stores keep L2 for the gather working set.
    const uint32_t eu = (uint32_t)e;
    #pragma unroll
    for (int q = 0; q < N_QOUT; ++q)
        __builtin_nontemporal_store(acc[q], ob[q] + eu);
}

extern "C" void kernel_launch(void* const* d_in, const int* in_sizes, int n_in,
                              void* d_out, int out_size, void* d_ws, size_t ws_size,
                              hipStream_t stream) {
    const float* fields  = (const float*)d_in[0];   // (12, N_CELLS) f32
    const int*   src_idx = (const int*)  d_in[1];   // (N_HALO, 2)  i32
    const float* weights = (const float*)d_in[2];   // (N_HALO, 2)  f32
    float*       out     = (float*)d_out;           // (11, N_HALO) f32

    dim3 grid(N_HALO / BLOCK);
    dim3 block(BLOCK);
    AI4SWE_49323404427329_halo_gather<<<grid, block, 0, stream>>>(
        fields, src_idx, weights, out);
}